// RNN_41927470744004
// MI455X (gfx1250) — compile-verified
//
#include <hip/hip_runtime.h>
#include <math.h>

// ---------------------------------------------------------------------------
// RNN forward on gfx1250:
//   Phase 1 (WMMA f32 16x16x4, 2x2 reg-blocked):  AW = x @ w0[0:512,:] + b0
//   Phase 2 (persistent, global barrier): for t: mem = sigmoid(AW[t] + mem @ w0[512:,:])
//            mem_t overwrites AW[t]  (history stored in place)
//   Phase 3 (WMMA f32 16x16x4, 2x2 reg-blocked):  out = AW @ w_out + b_out
// ---------------------------------------------------------------------------

typedef __attribute__((ext_vector_type(2))) float v2f;
typedef __attribute__((ext_vector_type(8))) float v8f;

#define T_STEPS   16384
#define SIZE_IN   512
#define SIZE_MEM  2048
#define SIZE_OUT  512
#define REC_BLOCKS 32           // persistent blocks in recurrence (must co-reside)

#define WMMA_F32(c, a, b) \
    __builtin_amdgcn_wmma_f32_16x16x4_f32(false, (a), false, (b), (short)0, (c), false, false)

// ---------------- generic 2x2 reg-blocked f32 WMMA GEMM tile ----------------
// Wave computes a 32(M) x 32(N) output tile of  D = A[MxK] @ B[KxN] + bias[N].
// A row-major lda=K, B row-major ldb=N, D row-major ldd=N.
template<int K>
__device__ __forceinline__ void gemm_tile_32x32(
    const float* __restrict__ A, const float* __restrict__ B,
    const float* __restrict__ bias, float* __restrict__ D,
    int ldaK, int ldbN, int m0, int n0, int lane)
{
    const int lh  = lane >> 4;         // 0: lanes 0-15, 1: lanes 16-31
    const int l15 = lane & 15;
    const int kk  = 2 * lh;            // K sub-offset {0,2} within each K-step

    const float* arow0 = A + (size_t)(m0 + l15)      * ldaK;
    const float* arow1 = A + (size_t)(m0 + 16 + l15) * ldaK;
    const int bc0 = n0 + l15;
    const int bc1 = n0 + 16 + l15;

    v8f c00 = {}, c01 = {}, c10 = {}, c11 = {};

    #pragma unroll 2
    for (int k0 = 0; k0 < K; k0 += 4) {
        // A fragments (16x4): lane holds A[row][k0+kk], A[row][k0+kk+1]
        v2f a0 = *(const v2f*)(arow0 + k0 + kk);
        v2f a1 = *(const v2f*)(arow1 + k0 + kk);
        // B fragments (4x16): lane holds B[k0+kk][col], B[k0+kk+1][col]
        const float* brow = B + (size_t)(k0 + kk) * ldbN;
        v2f b0, b1;
        b0.x = brow[bc0];            b0.y = brow[ldbN + bc0];
        b1.x = brow[bc1];            b1.y = brow[ldbN + bc1];

        c00 = WMMA_F32(c00, a0, b0);
        c01 = WMMA_F32(c01, a0, b1);
        c10 = WMMA_F32(c10, a1, b0);
        c11 = WMMA_F32(c11, a1, b1);
    }

    const float bias0 = bias[bc0];
    const float bias1 = bias[bc1];
    #pragma unroll
    for (int r = 0; r < 8; ++r) {
        int mrow0 = m0 + r + (lh << 3);        // M = r or r+8 within tile 0
        int mrow1 = mrow0 + 16;                // second M tile
        D[(size_t)mrow0 * ldbN + bc0] = c00[r] + bias0;
        D[(size_t)mrow0 * ldbN + bc1] = c01[r] + bias1;
        D[(size_t)mrow1 * ldbN + bc0] = c10[r] + bias0;
        D[(size_t)mrow1 * ldbN + bc1] = c11[r] + bias1;
    }
}

// ---------------- Phase 1: AW = x @ w0[:512,:] + b0 (M=16384,K=512,N=2048) --
// block: 256 thr = 8 waves; each wave one 32x32 tile; block tile 32M x 256N.
// grid: (T/32, 2048/256)
__global__ void __launch_bounds__(256)
rnn_pre_gemm(const float* __restrict__ x,     // [T, 512]
             const float* __restrict__ w0,    // [2560, 2048] (rows 0..511 used)
             const float* __restrict__ b0,    // [2048]
             float* __restrict__ AW)          // [T, 2048]
{
    const int lane = threadIdx.x & 31;
    const int wave = threadIdx.x >> 5;
    const int m0 = blockIdx.x * 32;
    const int n0 = blockIdx.y * 256 + wave * 32;
    gemm_tile_32x32<SIZE_IN>(x, w0, b0, AW, SIZE_IN, SIZE_MEM, m0, n0, lane);
}

// ---------------- Phase 3: out = Mem @ w_out + b_out (K=2048, N=512) --------
// grid: (T/32, 512/256)
__global__ void __launch_bounds__(256)
rnn_out_gemm(const float* __restrict__ Mem,   // [T, 2048] (AW after phase 2)
             const float* __restrict__ wout,  // [2048, 512]
             const float* __restrict__ bout,  // [512]
             float* __restrict__ out)         // [T, 512]
{
    const int lane = threadIdx.x & 31;
    const int wave = threadIdx.x >> 5;
    const int m0 = blockIdx.x * 32;
    const int n0 = blockIdx.y * 256 + wave * 32;
    gemm_tile_32x32<SIZE_MEM>(Mem, wout, bout, out, SIZE_MEM, SIZE_OUT, m0, n0, lane);
}

// ---------------- Phase 2: sequential recurrence ---------------------------
// 32 blocks x 256 threads, persistent. Block b owns columns [64b, 64b+64).
// thread: col_local = tid&63, kslice = tid>>6 (4 k-slices of 512).
__device__ __forceinline__ void global_barrier(unsigned* cnt, unsigned expected)
{
    __syncthreads();
    if (threadIdx.x == 0) {
        __threadfence();   // publish this block's stores at agent scope
        __hip_atomic_fetch_add(cnt, 1u, __ATOMIC_RELEASE, __HIP_MEMORY_SCOPE_AGENT);
        while (__hip_atomic_load(cnt, __ATOMIC_ACQUIRE, __HIP_MEMORY_SCOPE_AGENT)
               < expected) {
            __builtin_amdgcn_s_sleep(1);
        }
    }
    __syncthreads();
}

__global__ void __launch_bounds__(256)
rnn_recurrence(const float* __restrict__ w0,   // [2560, 2048]; rows 512.. = W_mem
               float* __restrict__ AW,         // in: pre-activation; out: mem history
               float* __restrict__ memA,       // [2048] double buffer (zeroed)
               float* __restrict__ memB,       // [2048] double buffer
               unsigned* __restrict__ bar)     // zeroed barrier counter
{
    __shared__ float mem_sh[SIZE_MEM];
    __shared__ float red[256];

    const int tid    = threadIdx.x;
    const int clocal = tid & 63;
    const int kslice = tid >> 6;                      // 0..3
    const int col    = blockIdx.x * 64 + clocal;      // owned neuron
    const float* wcol = w0 + (size_t)(SIZE_IN + kslice * 512) * SIZE_MEM + col;

    for (int t = 0; t < T_STEPS; ++t) {
        const float* memIn  = (t & 1) ? memB : memA;
        float*       memOut = (t & 1) ? memA : memB;

        // stage previous hidden state in LDS
        #pragma unroll
        for (int i = 0; i < SIZE_MEM / 256; ++i)
            mem_sh[tid + i * 256] = memIn[tid + i * 256];
        __syncthreads();

        // partial dot over this thread's 512-row k-slice (column `col` of W_mem)
        float acc = 0.f;
        const float* ms = mem_sh + kslice * 512;
        #pragma unroll 8
        for (int k = 0; k < 512; ++k)
            acc = fmaf(ms[k], wcol[(size_t)k * SIZE_MEM], acc);

        red[tid] = acc;
        __syncthreads();

        if (kslice == 0) {
            float s = red[clocal] + red[64 + clocal]
                    + red[128 + clocal] + red[192 + clocal];
            size_t idx = (size_t)t * SIZE_MEM + col;
            s += AW[idx];                       // + x_t @ w0x + b0
            float m = 1.0f / (1.0f + expf(-s)); // sigmoid
            memOut[col] = m;                    // next step's input
            AW[idx]     = m;                    // history (A[t] slot now dead)
        }
        __syncthreads();                        // red[] reuse safety

        global_barrier(bar, (unsigned)(REC_BLOCKS * (t + 1)));
    }
}

// ---------------------------------------------------------------------------
extern "C" void kernel_launch(void* const* d_in, const int* in_sizes, int n_in,
                              void* d_out, int out_size, void* d_ws, size_t ws_size,
                              hipStream_t stream)
{
    const float* x    = (const float*)d_in[0];   // [16384, 512]
    const float* w0   = (const float*)d_in[1];   // [2560, 2048]
    const float* b0   = (const float*)d_in[2];   // [1, 2048]
    const float* wout = (const float*)d_in[3];   // [2048, 512]
    const float* bout = (const float*)d_in[4];   // [1, 512]
    float* out = (float*)d_out;                  // [16384, 512]

    // Workspace layout (needs ~134.3 MB):
    //   AW   : T*2048 floats (pre-activation, then mem history in place)
    //   memA : 2048 floats   (double buffer, zero-initialized = mem_0)
    //   memB : 2048 floats
    //   bar  : barrier counter
    char* ws = (char*)d_ws;
    float*    AW   = (float*)ws;
    float*    memA = (float*)(ws + (size_t)T_STEPS * SIZE_MEM * sizeof(float));
    float*    memB = memA + SIZE_MEM;
    unsigned* bar  = (unsigned*)(memB + SIZE_MEM);

    // zero memA/memB + barrier counter (capture-safe async memset)
    hipMemsetAsync(memA, 0, (2 * SIZE_MEM) * sizeof(float) + 64, stream);

    // Phase 1: AW = x @ w0[:512,:] + b0
    dim3 g1(T_STEPS / 32, SIZE_MEM / 256);
    rnn_pre_gemm<<<g1, 256, 0, stream>>>(x, w0, b0, AW);

    // Phase 2: sequential recurrence (persistent, 32 co-resident blocks)
    rnn_recurrence<<<REC_BLOCKS, 256, 0, stream>>>(w0, AW, memA, memB, bar);

    // Phase 3: out = Mem @ w_out + b_out
    dim3 g3(T_STEPS / 32, SIZE_OUT / 256);
    rnn_out_gemm<<<g3, 256, 0, stream>>>(AW, wout, bout, out);
}